// DetectionIntentionLoss_29987461661361
// MI455X (gfx1250) — compile-verified
//
#include <hip/hip_runtime.h>
#include <hip/hip_bf16.h>
#include <stdint.h>
#include <stddef.h>

// ---------------------------------------------------------------------------
// DetectionIntentionLoss for MI455X (gfx1250).
// Memory-bound (≈65MB traffic, ~2.8us @ 23.3TB/s); no matmul structure, so
// WMMA is inapplicable. CDNA5 paths used: async global->LDS staging
// (ASYNCcnt + s_wait_asynccnt), global_prefetch_b8, wave32 reductions.
// ---------------------------------------------------------------------------

namespace {

constexpr int kB = 8;
constexpr int kN = 131072;
constexpr int kM = 48;
constexpr int kC = 8;
constexpr int kBlock = 256;           // 8 wave32s per block
constexpr int kBlocksX = kN / kBlock; // 512
constexpr int kNumBlk = kBlocksX * kB;

constexpr float kIouPos = 0.6f;
constexpr float kIouNeg = 0.45f;
constexpr float kEps = 1e-6f;
constexpr float kBeta = 1.0f / 9.0f;
constexpr float kAlpha = 0.25f;
constexpr float kClsW = 1.0f, kBoxW = 1.0f, kIntW = 0.5f;

} // namespace

// ---------------- CDNA5 async global->LDS staging ---------------------------
#if defined(__HIP_DEVICE_COMPILE__) && defined(__gfx1250__) && \
    __has_builtin(__builtin_amdgcn_global_load_async_to_lds_b32)
#define USE_ASYNC_LDS 1
#endif

#if defined(USE_ASYNC_LDS)
typedef __attribute__((address_space(1))) int GlobalI32;
typedef __attribute__((address_space(3))) int LdsI32;

__device__ __forceinline__ void g2l_async_b32(const void* g, void* l) {
  __builtin_amdgcn_global_load_async_to_lds_b32((GlobalI32*)g, (LdsI32*)l, 0, 0);
}
__device__ __forceinline__ void g2l_wait() {
#if __has_builtin(__builtin_amdgcn_s_wait_asynccnt)
  __builtin_amdgcn_s_wait_asynccnt(0);
#else
  asm volatile("s_wait_asynccnt 0" ::: "memory");
#endif
}
#endif

// ---------------- kernel 0: zero the force-match keys ------------------------
__global__ void init_ws_kernel(unsigned long long* __restrict__ force) {
  int t = blockIdx.x * blockDim.x + threadIdx.x;
  if (t < kB * kM) force[t] = 0ull;
}

// ---------------- kernel 1: per-GT column max (force matching) ---------------
// key = (float_bits(iou) << 32) | (0xFFFFFFFF - anchor_idx)
// iou >= 0 so float bit pattern is monotonic; complemented index makes
// atomicMax pick the SMALLEST anchor index on IoU ties (jnp.argmax semantics).
__global__ void __launch_bounds__(kBlock)
force_match_kernel(const float* __restrict__ anchors,
                   const float* __restrict__ gt_boxes,
                   unsigned long long* __restrict__ force) {
  const int b = blockIdx.y;
  const int t = threadIdx.x;
  const int n = blockIdx.x * kBlock + t; // anchor index within batch

  __shared__ float s_gt[kM * 5];
  __shared__ float s_an[kBlock * 5];
  __shared__ unsigned long long s_best[kM];

  const float* gsrc = gt_boxes + (size_t)b * kM * 5;
  const float* asrc = anchors + (size_t)blockIdx.x * kBlock * 5;

#if defined(USE_ASYNC_LDS)
  if (t < kM * 5) g2l_async_b32(gsrc + t, &s_gt[t]);
#pragma unroll
  for (int j = 0; j < 5; ++j)
    g2l_async_b32(asrc + t + kBlock * j, &s_an[t + kBlock * j]);
  g2l_wait();
#else
  if (t < kM * 5) s_gt[t] = gsrc[t];
#pragma unroll
  for (int j = 0; j < 5; ++j) s_an[t + kBlock * j] = asrc[t + kBlock * j];
#endif
  if (t < kM) s_best[t] = 0ull;
  __syncthreads();

  const float a0 = s_an[t * 5 + 0], a1 = s_an[t * 5 + 1];
  const float a2 = s_an[t * 5 + 2], a3 = s_an[t * 5 + 3];
  const float ahx = a2 * 0.5f, ahy = a3 * 0.5f;
  const float areaA = a2 * a3;
  const unsigned long long idbits =
      (unsigned long long)(0xFFFFFFFFu - (unsigned)n);
  const int rot = t % kM; // stagger to spread LDS-atomic contention

  for (int j = 0; j < kM; ++j) {
    int m = rot + j;
    if (m >= kM) m -= kM;
    const float* g = &s_gt[m * 5];
    const float ghx = g[2] * 0.5f, ghy = g[3] * 0.5f;
    float ix = fminf(a0 + ahx, g[0] + ghx) - fmaxf(a0 - ahx, g[0] - ghx);
    float iy = fminf(a1 + ahy, g[1] + ghy) - fmaxf(a1 - ahy, g[1] - ghy);
    ix = fmaxf(ix, 0.0f);
    iy = fmaxf(iy, 0.0f);
    const float inter = ix * iy;
    const float iou = inter / (areaA + g[2] * g[3] - inter + kEps);
    const unsigned long long key =
        ((unsigned long long)__float_as_uint(iou) << 32) | idbits;
    atomicMax(&s_best[m], key);
  }
  __syncthreads();
  if (t < kM) atomicMax(&force[b * kM + t], s_best[t]);
}

// ---------------- kernel 2: per-anchor losses -> per-block partials ----------
__global__ void __launch_bounds__(kBlock)
loss_kernel(const float* __restrict__ cls_logits,
            const float* __restrict__ box_preds,
            const float* __restrict__ intent_logits,
            const float* __restrict__ anchors,
            const float* __restrict__ gt_boxes,
            const int* __restrict__ gt_int,
            const unsigned long long* __restrict__ force,
            float4* __restrict__ partials) {
  const int b = blockIdx.y;
  const int t = threadIdx.x;
  const int n = blockIdx.x * kBlock + t;
  const size_t gid = (size_t)b * kN + n;

  __shared__ float s_gt[kM * 5];
  __shared__ int s_gi[kM];
  __shared__ float s_an[kBlock * 5];
  __shared__ unsigned s_fidx[kM];
  __shared__ int s_fok[kM];
  __shared__ float s_red[kBlock / 32][4];

  const float* gsrc = gt_boxes + (size_t)b * kM * 5;
  const int* isrc = gt_int + (size_t)b * kM;
  const float* asrc = anchors + (size_t)blockIdx.x * kBlock * 5;

#if defined(USE_ASYNC_LDS)
  if (t < kM * 5) g2l_async_b32(gsrc + t, &s_gt[t]);
  if (t < kM) g2l_async_b32(isrc + t, &s_gi[t]);
#pragma unroll
  for (int j = 0; j < 5; ++j)
    g2l_async_b32(asrc + t + kBlock * j, &s_an[t + kBlock * j]);
  g2l_wait();
#else
  if (t < kM * 5) s_gt[t] = gsrc[t];
  if (t < kM) s_gi[t] = isrc[t];
#pragma unroll
  for (int j = 0; j < 5; ++j) s_an[t + kBlock * j] = asrc[t + kBlock * j];
#endif
  if (t < kM) {
    const unsigned long long key = force[b * kM + t];
    s_fidx[t] = 0xFFFFFFFFu - (unsigned)key;
    s_fok[t] = (__uint_as_float((unsigned)(key >> 32)) >= kIouNeg) ? 1 : 0;
  }
  __syncthreads();

  // Prefetch the rows the positive-branch will touch (global_prefetch_b8).
  __builtin_prefetch(intent_logits + gid * kC, 0, 1);
  __builtin_prefetch(box_preds + gid * 6, 0, 1);

  const float a0 = s_an[t * 5 + 0], a1 = s_an[t * 5 + 1];
  const float a2 = s_an[t * 5 + 2], a3 = s_an[t * 5 + 3];
  const float a4 = s_an[t * 5 + 4];
  const float ahx = a2 * 0.5f, ahy = a3 * 0.5f;
  const float areaA = a2 * a3;

  // Row max/argmax; sequential m with strict '>' == jnp.argmax first-occurrence.
  float best = -1.0f;
  int bm = 0;
  for (int m = 0; m < kM; ++m) {
    const float* g = &s_gt[m * 5];
    const float ghx = g[2] * 0.5f, ghy = g[3] * 0.5f;
    float ix = fminf(a0 + ahx, g[0] + ghx) - fmaxf(a0 - ahx, g[0] - ghx);
    float iy = fminf(a1 + ahy, g[1] + ghy) - fmaxf(a1 - ahy, g[1] - ghy);
    ix = fmaxf(ix, 0.0f);
    iy = fmaxf(iy, 0.0f);
    const float inter = ix * iy;
    const float iou = inter / (areaA + g[2] * g[3] - inter + kEps);
    if (iou > best) { best = iou; bm = m; }
  }

  bool pos = best >= kIouPos;
  const bool neg = best < kIouNeg;
  bool forced = false;
  for (int m = 0; m < kM; ++m)
    if (s_fok[m] && s_fidx[m] == (unsigned)n) forced = true;
  pos = pos || forced;
  const bool validv = pos || neg;

  // ---- focal loss (on valid anchors) ----
  const float tcls = pos ? 1.0f : 0.0f;
  const float x = cls_logits[gid];
  const float ce = fmaxf(x, 0.0f) - x * tcls + log1pf(expf(-fabsf(x)));
  const float p = 1.0f / (1.0f + expf(-x));
  const float p_t = p * tcls + (1.0f - p) * (1.0f - tcls);
  const float alpha_t = kAlpha * tcls + (1.0f - kAlpha) * (1.0f - tcls);
  const float om = 1.0f - p_t;
  float csum = validv ? (alpha_t * ce * om * om) : 0.0f;

  float bsum = 0.0f, isum = 0.0f;
  float npv = pos ? 1.0f : 0.0f;

  if (pos) {
    // ---- smooth-L1 on box deltas vs the per-anchor best GT ----
    const float* ag = &s_gt[bm * 5];
    const float da = ag[4] - a4;
    float tgt6[6];
    tgt6[0] = (ag[0] - a0) / (a2 + kEps);
    tgt6[1] = (ag[1] - a1) / (a3 + kEps);
    tgt6[2] = logf(ag[2] / (a2 + kEps) + kEps);
    tgt6[3] = logf(ag[3] / (a3 + kEps) + kEps);
    tgt6[4] = sinf(da);
    tgt6[5] = cosf(da);
    const float* bp = box_preds + gid * 6;
#pragma unroll
    for (int k = 0; k < 6; ++k) {
      const float d = fabsf(bp[k] - tgt6[k]);
      bsum += (d < kBeta) ? (0.5f * d * d / kBeta) : (d - 0.5f * kBeta);
    }
    // ---- intention cross-entropy ----
    const float* il = intent_logits + gid * kC;
    float v[kC];
#pragma unroll
    for (int k = 0; k < kC; ++k) v[k] = il[k];
    float mx = v[0];
#pragma unroll
    for (int k = 1; k < kC; ++k) mx = fmaxf(mx, v[k]);
    float s = 0.0f;
#pragma unroll
    for (int k = 0; k < kC; ++k) s += expf(v[k] - mx);
    const float lse = mx + logf(s);
    int tg = s_gi[bm];
    tg = (tg < 0) ? 0 : ((tg > kC - 1) ? kC - 1 : tg);
    isum = lse - v[tg];
  }

  // ---- wave32 shuffle reduction, then cross-wave LDS reduction ----
  const int lane = t & 31, wid = t >> 5;
#pragma unroll
  for (int off = 16; off > 0; off >>= 1) {
    csum += __shfl_down(csum, off, 32);
    bsum += __shfl_down(bsum, off, 32);
    isum += __shfl_down(isum, off, 32);
    npv += __shfl_down(npv, off, 32);
  }
  if (lane == 0) {
    s_red[wid][0] = csum;
    s_red[wid][1] = bsum;
    s_red[wid][2] = isum;
    s_red[wid][3] = npv;
  }
  __syncthreads();
  if (t == 0) {
    float c = 0.0f, bb = 0.0f, ii = 0.0f, pp = 0.0f;
#pragma unroll
    for (int w = 0; w < kBlock / 32; ++w) {
      c += s_red[w][0];
      bb += s_red[w][1];
      ii += s_red[w][2];
      pp += s_red[w][3];
    }
    const int blk = blockIdx.y * gridDim.x + blockIdx.x;
    partials[blk] = make_float4(c, bb, ii, pp);
  }
}

// ---------------- kernel 3: deterministic final reduction --------------------
__global__ void __launch_bounds__(256)
finalize_kernel(const float4* __restrict__ part, float* __restrict__ out) {
  __shared__ float4 s[256];
  float c = 0.0f, b = 0.0f, i = 0.0f, p = 0.0f;
  for (int k = threadIdx.x; k < kNumBlk; k += 256) {
    const float4 v = part[k];
    c += v.x; b += v.y; i += v.z; p += v.w;
  }
  s[threadIdx.x] = make_float4(c, b, i, p);
  __syncthreads();
  for (int off = 128; off > 0; off >>= 1) {
    if (threadIdx.x < off) {
      const float4 o = s[threadIdx.x + off];
      const float4 m = s[threadIdx.x];
      s[threadIdx.x] = make_float4(m.x + o.x, m.y + o.y, m.z + o.z, m.w + o.w);
    }
    __syncthreads();
  }
  if (threadIdx.x == 0) {
    const float4 r = s[0];
    const float denom = fmaxf(1.0f, r.w);
    const float cls = r.x / denom;
    const float box = r.y / denom;
    const float intent = r.z / denom;
    out[0] = kClsW * cls + kBoxW * box + kIntW * intent;
    out[1] = cls;
    out[2] = box;
    out[3] = intent;
    out[4] = r.w;
  }
}

// ---------------------------------------------------------------------------
extern "C" void kernel_launch(void* const* d_in, const int* in_sizes, int n_in,
                              void* d_out, int out_size, void* d_ws,
                              size_t ws_size, hipStream_t stream) {
  const float* cls_logits = (const float*)d_in[0];
  const float* box_preds = (const float*)d_in[1];
  const float* intent_logits = (const float*)d_in[2];
  const float* anchors = (const float*)d_in[3];
  const float* gt_boxes = (const float*)d_in[4];
  const int* gt_int = (const int*)d_in[5];
  float* out = (float*)d_out;

  // workspace layout: [0,3072) force keys (384 u64), then 4096 float4 partials
  unsigned long long* force = (unsigned long long*)d_ws;
  float4* partials =
      (float4*)((char*)d_ws + (size_t)kB * kM * sizeof(unsigned long long));

  init_ws_kernel<<<1, 512, 0, stream>>>(force);

  dim3 grid(kBlocksX, kB);
  force_match_kernel<<<grid, kBlock, 0, stream>>>(anchors, gt_boxes, force);
  loss_kernel<<<grid, kBlock, 0, stream>>>(cls_logits, box_preds, intent_logits,
                                           anchors, gt_boxes, gt_int, force,
                                           partials);
  finalize_kernel<<<1, 256, 0, stream>>>(partials, out);
}